// GATPlanner_37151467110624
// MI455X (gfx1250) — compile-verified
//
#include <hip/hip_runtime.h>
#include <math.h>

#define DEV __device__ __forceinline__

constexpr int B_ = 2, H_ = 50, W_ = 50, N_ = 2500, P_ = 4, KMAX = 128;
constexpr float LRELU_ = 0.2f;
constexpr float ZTOL_ = 1e-9f;

typedef float v2f __attribute__((ext_vector_type(2)));
typedef float v8f __attribute__((ext_vector_type(8)));

#if defined(__has_builtin)
#if __has_builtin(__builtin_amdgcn_wmma_f32_16x16x4_f32)
#define HAVE_WMMA_F32 1
#endif
#endif

// ---------------------------------------------------------------- utilities
DEV float dev_sigmoid(float v) { return 1.0f / (1.0f + expf(-v)); }
DEV int dev_min(int a, int b) { return a < b ? a : b; }

// ---------------------------------------------------------------- sigmoid
__global__ void k_sigmoid(const float* __restrict__ x, float* __restrict__ y, int n) {
  int i = blockIdx.x * blockDim.x + threadIdx.x;
  if (i < n) y[i] = dev_sigmoid(x[i]);
}

// ------------------------------------------------- 3x3 conv, zero pad, relu
__global__ void k_conv_relu(const float* __restrict__ in, const float* __restrict__ w,
                            const float* __restrict__ bias, float* __restrict__ out,
                            int Cin, int Cout) {
  int idx = blockIdx.x * blockDim.x + threadIdx.x;
  int total = B_ * Cout * N_;
  if (idx >= total) return;
  int pix = idx % N_;
  int t = idx / N_;
  int oc = t % Cout;
  int b = t / Cout;
  int py = pix / W_, px = pix % W_;
  float acc = bias[oc];
  for (int ic = 0; ic < Cin; ++ic) {
    const float* ip = in + ((size_t)b * Cin + ic) * N_;
    const float* wp = w + ((size_t)oc * Cin + ic) * 9;
    for (int ky = 0; ky < 3; ++ky) {
      int yy = py + ky - 1;
      if (yy < 0 || yy >= H_) continue;
      for (int kx = 0; kx < 3; ++kx) {
        int xx = px + kx - 1;
        if (xx < 0 || xx >= W_) continue;
        acc += wp[ky * 3 + kx] * ip[yy * W_ + xx];
      }
    }
  }
  out[idx] = fmaxf(acc, 0.0f);
}

// ---------------------- 3x3 conv, pad value -999, relu, optional ch. scale
__global__ void k_conv_negpad_relu(const float* __restrict__ in, const float* __restrict__ att,
                                   const float* __restrict__ w, const float* __restrict__ bias,
                                   float* __restrict__ out, int Cin, int Cout) {
  int idx = blockIdx.x * blockDim.x + threadIdx.x;
  int total = B_ * Cout * N_;
  if (idx >= total) return;
  int pix = idx % N_;
  int t = idx / N_;
  int oc = t % Cout;
  int b = t / Cout;
  int py = pix / W_, px = pix % W_;
  float acc = bias[oc];
  for (int ic = 0; ic < Cin; ++ic) {
    float sc = att ? att[b * Cin + ic] : 1.0f;
    const float* ip = in + ((size_t)b * Cin + ic) * N_;
    const float* wp = w + ((size_t)oc * Cin + ic) * 9;
    for (int ky = 0; ky < 3; ++ky) {
      int yy = py + ky - 1;
      for (int kx = 0; kx < 3; ++kx) {
        int xx = px + kx - 1;
        bool inb = (yy >= 0 && yy < H_ && xx >= 0 && xx < W_);
        float iv = inb ? ip[yy * W_ + xx] * sc : -999.0f;
        acc += wp[ky * 3 + kx] * iv;
      }
    }
  }
  out[idx] = fmaxf(acc, 0.0f);
}

// ----------------------------------------- build compact per-row nnz lists
__global__ void k_csr(const float* __restrict__ S, int* __restrict__ cnt,
                      int* __restrict__ idx, float* __restrict__ val) {
  int row = blockIdx.x;  // (b*2+s)*N + i
  __shared__ int c;
  if (threadIdx.x == 0) c = 0;
  __syncthreads();
  const float* srow = S + (size_t)row * N_;
  for (int j = threadIdx.x; j < N_; j += blockDim.x) {
    float v = srow[j];
    if (fabsf(v) > ZTOL_) {
      int p = atomicAdd(&c, 1);
      if (p < KMAX) {
        idx[(size_t)row * KMAX + p] = j;
        val[(size_t)row * KMAX + p] = v;
      }
    }
  }
  __syncthreads();
  if (threadIdx.x == 0) cnt[row] = (c < KMAX) ? c : KMAX;
}

// --------------------------------------------- x (B,G,N) -> xt (B,N,G)
__global__ void k_transpose(const float* __restrict__ x, float* __restrict__ xt, int G) {
  int idx = blockIdx.x * blockDim.x + threadIdx.x;
  if (idx >= B_ * N_ * G) return;
  int g = idx % G;
  int t = idx / G;
  int n = t % N_;
  int b = t / N_;
  xt[idx] = x[((size_t)b * G + g) * N_ + n];
}

// ----------------------------- s1 = a[:, :G].x ; s2 = a[:, G:].x  (B,P,N)
__global__ void k_s12(const float* __restrict__ x, const float* __restrict__ a,
                      float* __restrict__ s1, float* __restrict__ s2, int G) {
  int idx = blockIdx.x * blockDim.x + threadIdx.x;
  if (idx >= B_ * P_ * N_) return;
  int i = idx % N_;
  int t = idx / N_;
  int p = t % P_;
  int b = t / P_;
  float u = 0.f, v = 0.f;
  for (int g = 0; g < G; ++g) {
    float xv = x[((size_t)b * G + g) * N_ + i];
    u += a[p * 2 * G + g] * xv;
    v += a[p * 2 * G + G + g] * xv;
  }
  s1[idx] = u;
  s2[idx] = v;
}

// ------------- sparse masked softmax attention + neighbor aggregation (z1)
// one 128-thread block per (b,p,node)
__global__ void k_attn_z1(const float* __restrict__ s1, const float* __restrict__ s2,
                          const float* __restrict__ xt, const int* __restrict__ cnt,
                          const int* __restrict__ idxs, const float* __restrict__ vals,
                          float* __restrict__ z1, int G, int sSel) {
  int node = blockIdx.x % N_;
  int t = blockIdx.x / N_;
  int p = t % P_;
  int b = t / P_;
  int tid = threadIdx.x;
  __shared__ float alpha[KMAX];
  __shared__ int jv[KMAX];
  __shared__ float red[128];

  int crow = (b * 2 + sSel) * N_ + node;
  int c = cnt[crow];
  float s1v = s1[(b * P_ + p) * N_ + node];
  float ev = -3.0e38f;
  if (tid < c) {
    int j = idxs[(size_t)crow * KMAX + tid];
    jv[tid] = j;
    float e = s1v + s2[(b * P_ + p) * N_ + j];
    ev = (e >= 0.0f) ? e : LRELU_ * e;
  }
  red[tid] = ev;
  __syncthreads();
  for (int s = 64; s > 0; s >>= 1) {
    if (tid < s) red[tid] = fmaxf(red[tid], red[tid + s]);
    __syncthreads();
  }
  float mx = red[0];
  __syncthreads();
  float ex = (tid < c) ? expf(ev - mx) : 0.0f;
  red[tid] = ex;
  __syncthreads();
  for (int s = 64; s > 0; s >>= 1) {
    if (tid < s) red[tid] += red[tid + s];
    __syncthreads();
  }
  float sum = red[0];
  if (tid < c) alpha[tid] = (ex / sum) * vals[(size_t)crow * KMAX + tid];
  __syncthreads();

  for (int g = tid; g < G; g += blockDim.x) {
    float acc = 0.0f;
    for (int k = 0; k < c; ++k)
      acc += alpha[k] * xt[((size_t)b * N_ + jv[k]) * G + g];
    z1[(((size_t)b * P_ + p) * G + g) * N_ + node] = acc;
  }
}

// ------------------------------------------------ per-head feature GEMM
// y[b, p*F+f, n] = bias[f] + sum_g x[b,g,n] W[p,0,g,f] + sum_g z1[b,p,g,n] W[p,1,g,f]
// GEMM: D(F x N) = A(F x 2G) * B(2G x N), tiled 16x16, K steps of 4 (fp32 WMMA).
// Out-of-range lanes CLAMP their indices (no zero-guard needed: D rows f>=F and
// cols n>=N are never stored, so garbage there is harmless) -> EXEC stays all-1s
// and addresses are pure affine streams the compiler can strength-reduce.
__global__ void k_gat_y(const float* __restrict__ x, const float* __restrict__ z1,
                        const float* __restrict__ W, const float* __restrict__ bias,
                        float* __restrict__ out, int G, int F, int accumulate) {
  const int NT = (N_ + 15) / 16;
  const int FT = (F + 15) / 16;
  int bid = blockIdx.x;
  int nt = bid % NT; bid /= NT;
  int ft = bid % FT; bid /= FT;
  int p = bid % P_;
  int b = bid / P_;

  int lane = threadIdx.x;  // wave32: one wave per block
  int half = lane >> 4;
  int l16 = lane & 15;
  int nbase = nt * 16, fbase = ft * 16;
  int nS = nbase + l16;                 // true column for the store
  int n = dev_min(nS, N_ - 1);          // clamped column for loads
  int fA = dev_min(fbase + l16, F - 1); // clamped A-row for loads

  // A = W[p] viewed as (2G x F), contiguous across both taps
  const float* pA = W + (size_t)p * 2 * G * F + fA;
  const float* pBx = x + (size_t)b * G * N_ + n;                 // rows 0..G-1
  const float* pBz = z1 + ((size_t)b * P_ + p) * (size_t)G * N_ + n;  // rows G..2G-1

  v8f acc = {0.f, 0.f, 0.f, 0.f, 0.f, 0.f, 0.f, 0.f};
#ifdef HAVE_WMMA_F32
  int kh = 2 * half;
#pragma unroll 4
  for (int k0 = 0; k0 < G; k0 += 4) {
    int ka = k0 + kh;
    v2f afrag, bfrag;
    afrag.x = pA[ka * F];
    afrag.y = pA[(ka + 1) * F];
    bfrag.x = pBx[ka * N_];
    bfrag.y = pBx[(ka + 1) * N_];
    acc = __builtin_amdgcn_wmma_f32_16x16x4_f32(false, afrag, false, bfrag,
                                                (short)0, acc, false, false);
  }
#pragma unroll 4
  for (int k0 = 0; k0 < G; k0 += 4) {
    int ka = k0 + kh;
    v2f afrag, bfrag;
    afrag.x = pA[(G + ka) * F];
    afrag.y = pA[(G + ka + 1) * F];
    bfrag.x = pBz[ka * N_];
    bfrag.y = pBz[(ka + 1) * N_];
    acc = __builtin_amdgcn_wmma_f32_16x16x4_f32(false, afrag, false, bfrag,
                                                (short)0, acc, false, false);
  }
#else
  // scalar fallback: each lane computes its 8 D entries directly
  for (int k = 0; k < G; ++k) {
    float bvx = pBx[k * N_];
    float bvz = pBz[k * N_];
    for (int r = 0; r < 8; ++r) {
      int f = dev_min(fbase + r + 8 * half, F - 1);
      const float* pAr = W + (size_t)p * 2 * G * F + f;
      acc[r] += pAr[k * F] * bvx + pAr[(G + k) * F] * bvz;
    }
  }
#endif
  for (int r = 0; r < 8; ++r) {
    int f = fbase + r + 8 * half;  // D VGPR r: lanes<16 -> M=r, lanes>=16 -> M=8+r
    if (f < F && nS < N_) {
      size_t oi = (((size_t)b * P_ + p) * F + f) * N_ + nS;
      float v = acc[r] + bias[f];
      if (accumulate)
        out[oi] += v;
      else
        out[oi] = v;
    }
  }
}

// --------------------------------------------------- channel max over N
__global__ void k_chmax(const float* __restrict__ d, float* __restrict__ att) {
  int bc = blockIdx.x;  // b*128 + c
  const float* row = d + (size_t)bc * N_;
  __shared__ float red[256];
  float m = -3.0e38f;
  for (int i = threadIdx.x; i < N_; i += 256) m = fmaxf(m, row[i]);
  red[threadIdx.x] = m;
  __syncthreads();
  for (int s = 128; s > 0; s >>= 1) {
    if (threadIdx.x < s) red[threadIdx.x] = fmaxf(red[threadIdx.x], red[threadIdx.x + s]);
    __syncthreads();
  }
  if (threadIdx.x == 0) att[bc] = red[0];
}

// --------------------------------------------- channel attention 128->128
__global__ void k_ca_mlp(const float* __restrict__ att, const float* __restrict__ w1,
                         const float* __restrict__ b1, const float* __restrict__ w2,
                         const float* __restrict__ b2, float* __restrict__ att2) {
  int b = blockIdx.x;
  int t = threadIdx.x;  // 128 threads
  __shared__ float av[128], m1[128];
  av[t] = att[b * 128 + t];
  __syncthreads();
  float acc = b1[t];
  for (int k = 0; k < 128; ++k) acc += w1[t * 128 + k] * av[k];
  m1[t] = fmaxf(acc, 0.0f);
  __syncthreads();
  float acc2 = b2[t];
  for (int k = 0; k < 128; ++k) acc2 += w2[t * 128 + k] * m1[k];
  att2[b * 128 + t] = dev_sigmoid(acc2);
}

// ----------------------------------------------------------- final MLP
__global__ void k_mlp1(const float* __restrict__ d2, const float* __restrict__ w1,
                       const float* __restrict__ b1, float* __restrict__ m) {
  int idx = blockIdx.x * blockDim.x + threadIdx.x;
  if (idx >= B_ * 512) return;
  int o = idx % 512;
  int b = idx / 512;
  float acc = b1[o];
  const float* wr = w1 + (size_t)o * N_;
  const float* xr = d2 + (size_t)b * N_;
  for (int k = 0; k < N_; ++k) acc += wr[k] * xr[k];
  m[idx] = fmaxf(acc, 0.0f);
}

__global__ void k_mlp2(const float* __restrict__ m, const float* __restrict__ w2,
                       const float* __restrict__ b2, float* __restrict__ out) {
  int idx = threadIdx.x;
  if (idx >= B_ * 5) return;
  int o = idx % 5;
  int b = idx / 5;
  float acc = b2[o];
  for (int k = 0; k < 512; ++k) acc += w2[o * 512 + k] * m[b * 512 + k];
  out[idx] = dev_sigmoid(acc);
}

// ================================================================= host
extern "C" void kernel_launch(void* const* d_in, const int* in_sizes, int n_in,
                              void* d_out, int out_size, void* d_ws, size_t ws_size,
                              hipStream_t stream) {
  (void)in_sizes; (void)n_in; (void)out_size; (void)ws_size;
  auto F = [&](int i) { return (const float*)d_in[i]; };

  const float* x = F(0);
  const float* Slist = F(1);
  const float* enc_w1 = F(2);  const float* enc_b1 = F(3);
  const float* enc_w2 = F(4);  const float* enc_b2 = F(5);
  const float* down0_a = F(6);  const float* down0_W = F(7);  const float* down0_b = F(8);
  const float* down1_a = F(9);  const float* down1_W = F(10); const float* down1_b = F(11);
  const float* up0_a = F(12);   const float* up0_W = F(13);   const float* up0_b = F(14);
  const float* up1_a = F(15);   const float* up1_W = F(16);   const float* up1_b = F(17);
  const float* sc0_a = F(18);   const float* sc0_W = F(19);   const float* sc0_b = F(20);
  const float* sc1_a = F(21);   const float* sc1_W = F(22);   const float* sc1_b = F(23);
  const float* ca_w1 = F(24);   const float* ca_b1 = F(25);
  const float* ca_w2 = F(26);   const float* ca_b2 = F(27);
  const float* dec_w1 = F(28);  const float* dec_b1 = F(29);
  const float* dec_w2 = F(30);  const float* dec_b2 = F(31);
  const float* mlp_w1 = F(32);  const float* mlp_b1 = F(33);
  const float* mlp_w2 = F(34);  const float* mlp_b2 = F(35);

  char* wsb = (char*)d_ws;
  size_t off = 0;
  auto AF = [&](size_t n) { float* p = (float*)(wsb + off); off += n * sizeof(float); return p; };
  auto AI = [&](size_t n) { int* p = (int*)(wsb + off); off += n * sizeof(int); return p; };

  float* hsig = AF((size_t)B_ * 3 * N_);
  float* e1   = AF((size_t)B_ * 32 * N_);
  float* f0   = AF((size_t)B_ * 32 * N_);
  float* f1   = AF((size_t)B_ * 64 * N_);
  float* f2   = AF((size_t)B_ * 32 * N_);
  float* g1   = AF((size_t)B_ * 64 * N_);
  float* g2   = AF((size_t)B_ * 128 * N_);
  float* xt   = AF((size_t)B_ * N_ * 64);
  float* s1a  = AF((size_t)B_ * P_ * N_);
  float* s2a  = AF((size_t)B_ * P_ * N_);
  float* z1   = AF((size_t)B_ * P_ * 64 * N_);
  float* att  = AF((size_t)B_ * 128);
  float* att2 = AF((size_t)B_ * 128);
  float* d1   = AF((size_t)B_ * 32 * N_);
  float* d2   = AF((size_t)B_ * N_);
  float* mlpm = AF((size_t)B_ * 512);
  float* csr_val = AF((size_t)B_ * 2 * N_ * KMAX);
  int* csr_idx = AI((size_t)B_ * 2 * N_ * KMAX);
  int* csr_cnt = AI((size_t)B_ * 2 * N_);

  const int THR = 256;
  auto blk = [&](long total) { return (int)((total + THR - 1) / THR); };

  // encoder
  k_sigmoid<<<blk((long)B_ * 3 * N_), THR, 0, stream>>>(x, hsig, B_ * 3 * N_);
  k_conv_relu<<<blk((long)B_ * 32 * N_), THR, 0, stream>>>(hsig, enc_w1, enc_b1, e1, 3, 32);
  k_conv_relu<<<blk((long)B_ * 32 * N_), THR, 0, stream>>>(e1, enc_w2, enc_b2, f0, 32, 32);

  // one pass over Slist -> compact neighbor lists (shared by all 6 GAT calls)
  k_csr<<<B_ * 2 * N_, 128, 0, stream>>>(Slist, csr_cnt, csr_idx, csr_val);

  auto run_gat = [&](const float* xin, int G, int Fh, const float* a, const float* Wf,
                     const float* bias, int sSel, float* yout, int accumulate) {
    k_transpose<<<blk((long)B_ * N_ * G), THR, 0, stream>>>(xin, xt, G);
    k_s12<<<blk((long)B_ * P_ * N_), THR, 0, stream>>>(xin, a, s1a, s2a, G);
    k_attn_z1<<<B_ * P_ * N_, 128, 0, stream>>>(s1a, s2a, xt, csr_cnt, csr_idx, csr_val,
                                                z1, G, sSel);
    int FT = (Fh + 15) / 16, NT = (N_ + 15) / 16;
    k_gat_y<<<B_ * P_ * FT * NT, 32, 0, stream>>>(xin, z1, Wf, bias, yout, G, Fh, accumulate);
  };

  run_gat(f0, 32, 16, down0_a, down0_W, down0_b, /*S0*/ 0, f1, 0);
  run_gat(f1, 64, 8,  down1_a, down1_W, down1_b, /*S1*/ 1, f2, 0);
  run_gat(f2, 32, 16, up0_a,   up0_W,   up0_b,   /*S1*/ 1, g1, 0);
  run_gat(f1, 64, 16, sc1_a,   sc1_W,   sc1_b,   /*S1*/ 1, g1, 1);
  run_gat(g1, 64, 32, up1_a,   up1_W,   up1_b,   /*S0*/ 0, g2, 0);
  run_gat(f0, 32, 32, sc0_a,   sc0_W,   sc0_b,   /*S0*/ 0, g2, 1);

  // channel attention
  k_chmax<<<B_ * 128, 256, 0, stream>>>(g2, att);
  k_ca_mlp<<<B_, 128, 0, stream>>>(att, ca_w1, ca_b1, ca_w2, ca_b2, att2);

  // decoder (attention scale folded into conv1 input read; -999 padding)
  k_conv_negpad_relu<<<blk((long)B_ * 32 * N_), THR, 0, stream>>>(g2, att2, dec_w1, dec_b1,
                                                                  d1, 128, 32);
  k_conv_negpad_relu<<<blk((long)B_ * 1 * N_), THR, 0, stream>>>(d1, nullptr, dec_w2, dec_b2,
                                                                 d2, 32, 1);

  // head MLP
  k_mlp1<<<blk((long)B_ * 512), THR, 0, stream>>>(d2, mlp_w1, mlp_b1, mlpm);
  k_mlp2<<<1, 32, 0, stream>>>(mlpm, mlp_w2, mlp_b2, (float*)d_out);
}